// GCL_47467978556197
// MI455X (gfx1250) — compile-verified
//
#include <hip/hip_runtime.h>

typedef __attribute__((ext_vector_type(2))) float v2f;
typedef __attribute__((ext_vector_type(8))) float v8f;

#define IN_DIM  512
#define HID_DIM 256
#define OUT_DIM 256
#define NCOL    256   // all three GEMMs produce 256 columns

static __device__ __forceinline__ v8f wmma_f32_k4(v2f a, v2f b, v8f c) {
    // D = A(16x4, f32) * B(4x16, f32) + C(16x16, f32)
    return __builtin_amdgcn_wmma_f32_16x16x4_f32(
        /*neg_a=*/false, a, /*neg_b=*/false, b,
        /*c_mod=*/(short)0, c, /*reuse_a=*/false, /*reuse_b=*/false);
}

// ---------------- degree / normalization ----------------

__global__ void deg_init_kernel(float* deg, int n) {
    int i = blockIdx.x * blockDim.x + threadIdx.x;
    if (i < n) deg[i] = 1.0f;   // self-loop contribution
}

__global__ void deg_count_kernel(const int* __restrict__ dst, float* deg, int e) {
    int i = blockIdx.x * blockDim.x + threadIdx.x;
    if (i < e) atomicAdd(&deg[dst[i]], 1.0f);
}

__global__ void deg_rsqrt_kernel(float* deg, int n) {
    int i = blockIdx.x * blockDim.x + threadIdx.x;
    if (i < n) deg[i] = rsqrtf(deg[i]);   // dis = deg^{-1/2}
}

// ---------------- WMMA f32 GEMM: C[M,256] = A[M,K] @ B[K,256] (+bias) ----------------
// Block: 256 threads = 8 waves. Each wave: 16 rows x 64 cols. Block: 128 x 64.
__global__ __launch_bounds__(256)
void gemm_wmma_f32(const float* __restrict__ A, const float* __restrict__ B,
                   const float* __restrict__ bias, float* __restrict__ C,
                   int M, int K) {
    const int lane = threadIdx.x & 31;
    const int wave = threadIdx.x >> 5;
    const int half = lane >> 4;      // 0: lanes 0-15, 1: lanes 16-31
    const int l15  = lane & 15;

    const int m0 = (blockIdx.x * 8 + wave) * 16;
    const int n0 = blockIdx.y * 64;
    if (m0 >= M) return;             // wave-uniform exit: EXEC stays all-ones for WMMA

    v8f acc0 = {}, acc1 = {}, acc2 = {}, acc3 = {};

    // Clamp out-of-range rows to row 0: their (garbage) outputs are store-guarded,
    // and valid rows never mix with them (WMMA row m depends only on A row m).
    int mrow = m0 + l15;
    if (mrow >= M) mrow = 0;

    // A: lane walks A[mrow, k + 2*half .. +1] ; step 4 per iteration
    const float* ap = A + (size_t)mrow * K + 2 * half;
    // B: lane reads rows (k + 2*half) and (k + 2*half + 1), col n0 + l15
    const float* bp = B + (size_t)(2 * half) * NCOL + n0 + l15;

    for (int k = 0; k < K; k += 4) {
        v2f a = *(const v2f*)ap;     // global_load_b64, 8B aligned (K even)
        ap += 4;
        v2f b;
        b.x = bp[0];           b.y = bp[NCOL];        acc0 = wmma_f32_k4(a, b, acc0);
        b.x = bp[16];          b.y = bp[NCOL + 16];   acc1 = wmma_f32_k4(a, b, acc1);
        b.x = bp[32];          b.y = bp[NCOL + 32];   acc2 = wmma_f32_k4(a, b, acc2);
        b.x = bp[48];          b.y = bp[NCOL + 48];   acc3 = wmma_f32_k4(a, b, acc3);
        bp += 4 * NCOL;
    }

    float bv0 = 0.0f, bv1 = 0.0f, bv2 = 0.0f, bv3 = 0.0f;
    if (bias) {
        bv0 = bias[n0 + 0  + l15];
        bv1 = bias[n0 + 16 + l15];
        bv2 = bias[n0 + 32 + l15];
        bv3 = bias[n0 + 48 + l15];
    }

    // C/D layout: VGPR r -> row m0+r (lanes 0-15) / m0+8+r (lanes 16-31)
    #pragma unroll
    for (int r = 0; r < 8; ++r) {
        int mm = m0 + 8 * half + r;
        if (mm < M) {
            float* cp = C + (size_t)mm * NCOL + n0 + l15;
            cp[0]  = acc0[r] + bv0;
            cp[16] = acc1[r] + bv1;
            cp[32] = acc2[r] + bv2;
            cp[48] = acc3[r] + bv3;
        }
    }
}

// ---------------- zero a float buffer ----------------

__global__ void zero_kernel(float* p, size_t n) {
    size_t i = (size_t)blockIdx.x * blockDim.x + threadIdx.x;
    if (i < n) p[i] = 0.0f;
}

// ---------------- edge scatter: agg[dst] += dis[src]*dis[dst]*h[src] ----------------
// One wave per edge; lanes stride the 256-wide feature row (L2-resident gathers).
__global__ __launch_bounds__(256)
void scatter_kernel(const int* __restrict__ src, const int* __restrict__ dst,
                    const float* __restrict__ dis, const float* __restrict__ h,
                    float* __restrict__ agg, int E, int F) {
    int e = blockIdx.x * 8 + (threadIdx.x >> 5);
    if (e >= E) return;
    int lane = threadIdx.x & 31;
    int s = src[e], d = dst[e];
    float c = dis[s] * dis[d];
    const float* hrow = h + (size_t)s * F;
    float* arow = agg + (size_t)d * F;
    for (int f = lane; f < F; f += 32)
        atomicAdd(&arow[f], c * hrow[f]);
}

// ---------------- combine: h[n,f] = (relu?)(agg + dis[n]^2 * h + b[f]) in place ----------------

__global__ void combine_kernel(const float* __restrict__ agg, float* __restrict__ h,
                               const float* __restrict__ dis, const float* __restrict__ b,
                               int n, int F, int do_relu) {
    size_t i = (size_t)blockIdx.x * blockDim.x + threadIdx.x;
    size_t total = (size_t)n * F;
    if (i >= total) return;
    int node = (int)(i / F);
    int f    = (int)(i % F);
    float di = dis[node];
    float v = agg[i] + di * di * h[i] + b[f];
    h[i] = do_relu ? fmaxf(v, 0.0f) : v;
}

// ---------------- launch ----------------

extern "C" void kernel_launch(void* const* d_in, const int* in_sizes, int n_in,
                              void* d_out, int out_size, void* d_ws, size_t ws_size,
                              hipStream_t stream) {
    const float* x  = (const float*)d_in[0];
    const int*   ei = (const int*)d_in[1];   // [2, E]: row 0 = src, row 1 = dst
    const float* W1 = (const float*)d_in[2];
    const float* b1 = (const float*)d_in[3];
    const float* W2 = (const float*)d_in[4];
    const float* b2 = (const float*)d_in[5];
    const float* Wp = (const float*)d_in[6];
    const float* bp = (const float*)d_in[7];
    float* out = (float*)d_out;

    const int N = in_sizes[0] / IN_DIM;     // 50000
    const int E = in_sizes[1] / 2;          // 800000
    const int* src = ei;
    const int* dst = ei + E;

    // workspace layout: dis[N] | buf1[N*256] | buf2[N*256]
    char* w = (char*)d_ws;
    size_t disBytes = (((size_t)N * 4) + 255) & ~(size_t)255;
    float* dis  = (float*)w;
    float* buf1 = (float*)(w + disBytes);
    float* buf2 = buf1 + (size_t)N * HID_DIM;

    const int T = 256;
    size_t NF = (size_t)N * HID_DIM;
    int nfBlocks = (int)((NF + T - 1) / T);

    // 1) dis = rsqrt(deg + 1)
    deg_init_kernel<<<(N + T - 1) / T, T, 0, stream>>>(dis, N);
    deg_count_kernel<<<(E + T - 1) / T, T, 0, stream>>>(dst, dis, E);
    deg_rsqrt_kernel<<<(N + T - 1) / T, T, 0, stream>>>(dis, N);

    // 2) h1 = x @ W1   [N,512]x[512,256] -> buf1
    {
        dim3 grid((N + 127) / 128, NCOL / 64);
        gemm_wmma_f32<<<grid, T, 0, stream>>>(x, W1, nullptr, buf1, N, IN_DIM);
    }
    // 3) agg1 = scatter(h1) -> buf2 ; buf1 = relu(agg1 + dis^2*h1 + b1)
    zero_kernel<<<nfBlocks, T, 0, stream>>>(buf2, NF);
    scatter_kernel<<<(E + 7) / 8, T, 0, stream>>>(src, dst, dis, buf1, buf2, E, HID_DIM);
    combine_kernel<<<nfBlocks, T, 0, stream>>>(buf2, buf1, dis, b1, N, HID_DIM, 1);

    // 4) h2 = buf1 @ W2 -> buf2
    {
        dim3 grid((N + 127) / 128, NCOL / 64);
        gemm_wmma_f32<<<grid, T, 0, stream>>>(buf1, W2, nullptr, buf2, N, HID_DIM);
    }
    // 5) agg2 = scatter(h2) -> buf1 ; buf2 = agg2 + dis^2*h2 + b2
    zero_kernel<<<nfBlocks, T, 0, stream>>>(buf1, NF);
    scatter_kernel<<<(E + 7) / 8, T, 0, stream>>>(src, dst, dis, buf2, buf1, E, OUT_DIM);
    combine_kernel<<<nfBlocks, T, 0, stream>>>(buf1, buf2, dis, b2, N, OUT_DIM, 0);

    // 6) out = buf2 @ Wp + bp
    {
        dim3 grid((N + 127) / 128, NCOL / 64);
        gemm_wmma_f32<<<grid, T, 0, stream>>>(buf2, Wp, bp, out, N, OUT_DIM);
    }
}